// BetaMPERLGraphConvLayer_73143293050932
// MI455X (gfx1250) — compile-verified
//
#include <hip/hip_runtime.h>
#include <math.h>

typedef __attribute__((ext_vector_type(2))) float v2f;
typedef __attribute__((ext_vector_type(8))) float v8f;

#define DIN   64
#define DOUT  64
#define RREL  8
#define NBASES 4
#define KDIM  (RREL * DIN)   // 512
#define EPSV  1e-8f
#define SHIFTV 1.01f
#define KCHUNK 64
#define LDSPITCH 66          // 64 + 2 pad: stride mod 64banks = 2 -> conflict-free-ish

// ---------------- zero workspace ----------------
__global__ void zero_f32(float* __restrict__ p, long n) {
  long i = (long)blockIdx.x * blockDim.x + threadIdx.x;
  long stride = (long)gridDim.x * blockDim.x;
  for (; i < n; i += stride) p[i] = 0.0f;
}

// ---------------- degree (row-sum of adjacency) ----------------
__global__ void deg_kernel(const int* __restrict__ rows, float* __restrict__ deg,
                           int N, int E) {
  int idx = blockIdx.x * blockDim.x + threadIdx.x;
  if (idx >= RREL * E) return;
  int r = idx / E;
  int row = rows[idx];
  atomicAdd(&deg[(long)r * N + row], 1.0f);
}

// ---------------- scatter: tmp[row, r*64+i] += X[col,i] / deg ----------------
__global__ void scatter_kernel(const float* __restrict__ X,
                               const int* __restrict__ rows,
                               const int* __restrict__ cols,
                               const float* __restrict__ deg,
                               float* __restrict__ tmp, int N, int E) {
  long t = (long)blockIdx.x * blockDim.x + threadIdx.x;
  long edge = t >> 6;            // 64 threads per edge, one feature each
  int  i    = (int)(t & 63);
  if (edge >= (long)RREL * E) return;
  int r   = (int)(edge / E);
  int row = rows[edge];
  int col = cols[edge];
  float inv = 1.0f / (deg[(long)r * N + row] + EPSV);
  float xv = X[(long)col * DIN + i];
  xv = (xv != xv) ? 0.0f : xv;   // nan_to_num
  atomicAdd(&tmp[(long)row * KDIM + r * DIN + i], xv * inv);
}

// ---------------- basis-combined weights, stored transposed [DOUT][KDIM] ----
__global__ void weight_kernel(const float* __restrict__ w_rel,
                              const float* __restrict__ w_bases,
                              float* __restrict__ wT) {
  int idx = blockIdx.x * blockDim.x + threadIdx.x;   // idx = k*64 + n
  if (idx >= KDIM * DOUT) return;
  int k = idx >> 6;
  int n = idx & 63;
  int r = k >> 6;
  int i = k & 63;
  float acc = 0.0f;
#pragma unroll
  for (int b = 0; b < NBASES; ++b)
    acc += w_rel[r * NBASES + b] * w_bases[((long)b * DIN + i) * DOUT + n];
  wT[(long)n * KDIM + k] = acc;  // transposed: [n][k]
}

// ---------------- WMMA GEMM + relu/bias/softplus epilogue ----------------
__global__ __launch_bounds__(128) void gemm_kernel(
    const float* __restrict__ tmp, const float* __restrict__ wTa,
    const float* __restrict__ wTb, const float* __restrict__ bias_a,
    const float* __restrict__ bias_b, float* __restrict__ out, int N) {
  __shared__ __align__(16) float lwa[DOUT][LDSPITCH];
  __shared__ __align__(16) float lwb[DOUT][LDSPITCH];

  const int tid    = threadIdx.x;
  const int lane   = tid & 31;
  const int wave   = tid >> 5;
  const int lane16 = lane & 15;
  const int koff   = (lane < 16) ? 0 : 2;   // K-offset per lane half (ISA layout)
  const int ntiles = (N + 15) >> 4;
  const int tile   = blockIdx.x * 4 + wave;
  const bool valid = (tile < ntiles);

  int row = tile * 16 + lane16;
  if (row >= N) row = N - 1;                // clamp loads; stores are guarded

  const v8f vzero = {0.f, 0.f, 0.f, 0.f, 0.f, 0.f, 0.f, 0.f};
  v8f acc_a[4], acc_b[4];
#pragma unroll
  for (int ct = 0; ct < 4; ++ct) { acc_a[ct] = vzero; acc_b[ct] = vzero; }

  for (int chunk = 0; chunk < KDIM / KCHUNK; ++chunk) {
    const int kbase = chunk * KCHUNK;
    __syncthreads();
    // stage both weight chunks into LDS (contiguous reads, padded pitch)
    for (int idx = tid; idx < DOUT * KCHUNK; idx += 128) {
      int n  = idx >> 6;
      int kl = idx & 63;
      lwa[n][kl] = wTa[(long)n * KDIM + kbase + kl];
      lwb[n][kl] = wTb[(long)n * KDIM + kbase + kl];
    }
    __syncthreads();
    if (!valid) continue;
#pragma unroll
    for (int ks = 0; ks < KCHUNK / 4; ++ks) {
      const int k0 = ks * 4;
      // A fragment 16x4: lanes 0-15 -> (K0,K1), lanes 16-31 -> (K2,K3)
      v2f a = *(const v2f*)&tmp[(long)row * KDIM + kbase + k0 + koff];
#pragma unroll
      for (int ct = 0; ct < 4; ++ct) {
        v2f ba = *(const v2f*)&lwa[ct * 16 + lane16][k0 + koff];
        v2f bb = *(const v2f*)&lwb[ct * 16 + lane16][k0 + koff];
        acc_a[ct] = __builtin_amdgcn_wmma_f32_16x16x4_f32(
            false, a, false, ba, (short)0, acc_a[ct], false, false);
        acc_b[ct] = __builtin_amdgcn_wmma_f32_16x16x4_f32(
            false, a, false, bb, (short)0, acc_b[ct], false, false);
      }
    }
  }
  if (!valid) return;

  // epilogue: relu -> +bias -> softplus -> +SHIFT
  const int mbase = tile * 16 + ((lane < 16) ? 0 : 8);
#pragma unroll
  for (int ct = 0; ct < 4; ++ct) {
    const int colc = ct * 16 + lane16;
    const float bal = bias_a[colc];
    const float bbl = bias_b[colc];
#pragma unroll
    for (int v = 0; v < 8; ++v) {
      int rrow = mbase + v;
      if (rrow < N) {
        float xa = fmaxf(acc_a[ct][v], 0.0f) + bal;
        float xb = fmaxf(acc_b[ct][v], 0.0f) + bbl;
        float spa = fmaxf(xa, 0.0f) + log1pf(__expf(-fabsf(xa)));
        float spb = fmaxf(xb, 0.0f) + log1pf(__expf(-fabsf(xb)));
        out[(long)rrow * DOUT + colc] = SHIFTV + spa;
        out[(long)N * DOUT + (long)rrow * DOUT + colc] = SHIFTV + spb;
      }
    }
  }
}

extern "C" void kernel_launch(void* const* d_in, const int* in_sizes, int n_in,
                              void* d_out, int out_size, void* d_ws, size_t ws_size,
                              hipStream_t stream) {
  const float* X      = (const float*)d_in[0];
  const int*   rows   = (const int*)d_in[1];
  const int*   cols   = (const int*)d_in[2];
  const float* wba    = (const float*)d_in[3];
  const float* wra    = (const float*)d_in[4];
  const float* wbb    = (const float*)d_in[5];
  const float* wrb    = (const float*)d_in[6];
  const float* bias_a = (const float*)d_in[7];
  const float* bias_b = (const float*)d_in[8];
  float* out = (float*)d_out;

  const int N = in_sizes[0] / DIN;     // 50000
  const int E = in_sizes[1] / RREL;    // 400000

  // workspace layout (floats): tmp[N*KDIM] | deg[R*N] | wTa[KDIM*DOUT] | wTb[...]
  float* ws  = (float*)d_ws;
  float* tmp = ws;
  float* deg = tmp + (long)N * KDIM;
  float* wTa = deg + (long)RREL * N;
  float* wTb = wTa + (long)KDIM * DOUT;

  long nz = (long)N * KDIM + (long)RREL * N;
  zero_f32<<<2048, 256, 0, stream>>>(ws, nz);

  {
    int total = RREL * E;
    deg_kernel<<<(total + 255) / 256, 256, 0, stream>>>(rows, deg, N, E);
  }
  {
    long total  = (long)RREL * E * 64;
    long blocks = (total + 255) / 256;
    scatter_kernel<<<(unsigned)blocks, 256, 0, stream>>>(X, rows, cols, deg, tmp, N, E);
  }
  weight_kernel<<<(KDIM * DOUT + 255) / 256, 256, 0, stream>>>(wra, wba, wTa);
  weight_kernel<<<(KDIM * DOUT + 255) / 256, 256, 0, stream>>>(wrb, wbb, wTb);
  {
    int ntiles = (N + 15) / 16;
    int blocks = (ntiles + 3) / 4;
    gemm_kernel<<<blocks, 128, 0, stream>>>(tmp, wTa, wTb, bias_a, bias_b, out, N);
  }
}